// HomoGCL_35699768164929
// MI455X (gfx1250) — compile-verified
//
#include <hip/hip_runtime.h>
#include <cstdint>
#include <cstddef>

#define HID 128

typedef float v2f __attribute__((ext_vector_type(2)));
typedef float v8f __attribute__((ext_vector_type(8)));

#if defined(__has_builtin)
# if __has_builtin(__builtin_amdgcn_wmma_f32_16x16x4_f32)
#  define HAVE_F32_WMMA 1
# endif
#endif
#ifndef HAVE_F32_WMMA
# define HAVE_F32_WMMA 0
#endif

// ---------------------------------------------------------------- utilities

__global__ void fill_zero_kernel(float* __restrict__ p, size_t n) {
  size_t i = (size_t)blockIdx.x * blockDim.x + threadIdx.x;
  size_t stride = (size_t)gridDim.x * blockDim.x;
  for (; i < n; i += stride) p[i] = 0.0f;
}

// edges: int64, shape [2, nE] row-major (row 0 = src, row 1 = dst)
__global__ void degree_kernel(const long long* __restrict__ e,
                              float* __restrict__ deg_out,
                              float* __restrict__ deg_in, int nE) {
  int i = blockIdx.x * blockDim.x + threadIdx.x;
  if (i < nE) {
    atomicAdd(&deg_out[(int)e[i]], 1.0f);
    atomicAdd(&deg_in[(int)e[(size_t)nE + i]], 1.0f);
  }
}

// d = (max(d,1))^{-1/2}
__global__ void rsqrt_clip_kernel(float* __restrict__ d, int n) {
  int i = blockIdx.x * blockDim.x + threadIdx.x;
  if (i < n) d[i] = rsqrtf(fmaxf(d[i], 1.0f));
}

// ---------------------------------------------------------------- dense GEMM
// Y[m, n] = (sum_k X[m,k] * W[k,n]) * rscale[m];  N fixed at 128.
// One 256-thread block = 8 waves; block handles a 16-row strip, wave w handles
// output columns [16w, 16w+16). K stepped by 4 with V_WMMA_F32_16X16X4_F32.

#if HAVE_F32_WMMA
template <int K>
__global__ __launch_bounds__(256) void gemm_rowscale_wmma(
    const float* __restrict__ X, const float* __restrict__ W,
    const float* __restrict__ rscale, float* __restrict__ Y) {
  const int lane = threadIdx.x & 31;
  const int wave = threadIdx.x >> 5;   // n-tile index 0..7
  const int m0 = blockIdx.x * 16;
  const int n0 = wave * 16;
  const int h  = lane >> 4;            // half-wave: 0 or 1
  const int lm = lane & 15;

  v8f acc = {0.f, 0.f, 0.f, 0.f, 0.f, 0.f, 0.f, 0.f};
  const float* arow = X + (size_t)(m0 + lm) * K;   // A: lane lm+16h holds row lm, K=k+2h..k+2h+1
  for (int k = 0; k < K; k += 4) {
    v2f a = *(const v2f*)(arow + k + 2 * h);
    v2f b;                                          // B: lane lm+16h holds col lm, K=k+2h..k+2h+1
    b.x = W[(size_t)(k + 2 * h) * HID + n0 + lm];
    b.y = W[(size_t)(k + 2 * h + 1) * HID + n0 + lm];
    acc = __builtin_amdgcn_wmma_f32_16x16x4_f32(
        /*neg_a=*/false, a, /*neg_b=*/false, b,
        /*c_mod=*/(short)0, acc, /*reuse_a=*/false, /*reuse_b=*/false);
  }
#pragma unroll
  for (int r = 0; r < 8; ++r) {                     // D: vgpr r, lane lm+16h -> row r+8h, col lm
    int row = m0 + r + 8 * h;
    Y[(size_t)row * HID + n0 + lm] = acc[r] * rscale[row];
  }
}
#else
template <int K>
__global__ void gemm_rowscale_fallback(const float* __restrict__ X,
                                       const float* __restrict__ W,
                                       const float* __restrict__ rscale,
                                       float* __restrict__ Y, int M) {
  int idx = blockIdx.x * blockDim.x + threadIdx.x;  // over M*HID
  if (idx >= M * HID) return;
  int row = idx >> 7, col = idx & (HID - 1);
  float s = 0.f;
  for (int k = 0; k < K; ++k) s += X[(size_t)row * K + k] * W[(size_t)k * HID + col];
  Y[idx] = s * rscale[row];
}
#endif

// ---------------------------------------------------------------- SpMM
// agg[dst] += h[src] : one wave per edge, each lane moves a float4 (128 cols).
__global__ void spmm_scatter_kernel(const long long* __restrict__ e,
                                    const float* __restrict__ h,
                                    float* __restrict__ agg, int nE) {
  const int lane = threadIdx.x & 31;
  const int wid = (int)(((size_t)blockIdx.x * blockDim.x + threadIdx.x) >> 5);
  if (wid >= nE) return;
  const int s = (int)e[wid];
  const int d = (int)e[(size_t)nE + wid];
  const float4 v = *(const float4*)(h + (size_t)s * HID + lane * 4);
  float* dst = agg + (size_t)d * HID + lane * 4;
  atomicAdd(dst + 0, v.x);
  atomicAdd(dst + 1, v.y);
  atomicAdd(dst + 2, v.z);
  atomicAdd(dst + 3, v.w);
}

// a[row,col] = (a[row,col] * din[row] + bias[col])  (optionally ReLU), in place
template <bool RELU>
__global__ void finish_kernel(float* __restrict__ a, const float* __restrict__ din,
                              const float* __restrict__ bias, int M) {
  int idx = blockIdx.x * blockDim.x + threadIdx.x;
  if (idx >= M * HID) return;
  int row = idx >> 7, col = idx & (HID - 1);
  float v = a[idx] * din[row] + bias[col];
  if (RELU) v = fmaxf(v, 0.f);
  a[idx] = v;
}

// ---------------------------------------------------------------- driver

extern "C" void kernel_launch(void* const* d_in, const int* in_sizes, int n_in,
                              void* d_out, int out_size, void* d_ws, size_t ws_size,
                              hipStream_t stream) {
  const int IN_DIM = 256;
  const int nE = in_sizes[0] / 2;          // 1,600,000
  const int M  = in_sizes[1] / IN_DIM;     // 100,000 (multiple of 16)

  const long long* edges[3] = {(const long long*)d_in[0],
                               (const long long*)d_in[2],
                               (const long long*)d_in[4]};
  const float* feats[3] = {(const float*)d_in[1], (const float*)d_in[3],
                           (const float*)d_in[5]};
  const float* W0 = (const float*)d_in[6];
  const float* b0 = (const float*)d_in[7];
  const float* W1 = (const float*)d_in[8];
  const float* b1 = (const float*)d_in[9];
  float* out = (float*)d_out;

  // workspace layout
  float* deg   = (float*)d_ws;                      // deg_out[M] then deg_in[M]
  float* bufA  = deg + 2 * (size_t)M;               // [M,128] GEMM output
  float* bufB  = bufA + (size_t)M * HID;            // [M,128] agg / h

  const int T = 256;
  const size_t MH = (size_t)M * HID;
  const int blocksMH = (int)((MH + T - 1) / T);
  const int blocksE  = (nE + T - 1) / T;
  const int blocksSp = (int)(((size_t)nE * 32 + T - 1) / T);
  const int mtiles   = M / 16;

  for (int g = 0; g < 3; ++g) {
    // degree normalizations for this graph
    fill_zero_kernel<<<(2 * M + T - 1) / T, T, 0, stream>>>(deg, 2 * (size_t)M);
    degree_kernel<<<blocksE, T, 0, stream>>>(edges[g], deg, deg + M, nE);
    rsqrt_clip_kernel<<<(2 * M + T - 1) / T, T, 0, stream>>>(deg, 2 * M);

    // ---- layer 1: h = relu(Din^-1/2 A Dout^-1/2 (x @ W0) + b0)
#if HAVE_F32_WMMA
    gemm_rowscale_wmma<256><<<mtiles, 256, 0, stream>>>(feats[g], W0, deg, bufA);
#else
    gemm_rowscale_fallback<256><<<blocksMH, T, 0, stream>>>(feats[g], W0, deg, bufA, M);
#endif
    fill_zero_kernel<<<blocksMH, T, 0, stream>>>(bufB, MH);
    spmm_scatter_kernel<<<blocksSp, T, 0, stream>>>(edges[g], bufA, bufB, nE);
    finish_kernel<true><<<blocksMH, T, 0, stream>>>(bufB, deg + M, b0, M);

    // ---- layer 2: z = Din^-1/2 A Dout^-1/2 (h @ W1) + b1  (scatter into d_out)
#if HAVE_F32_WMMA
    gemm_rowscale_wmma<128><<<mtiles, 256, 0, stream>>>(bufB, W1, deg, bufA);
#else
    gemm_rowscale_fallback<128><<<blocksMH, T, 0, stream>>>(bufB, W1, deg, bufA, M);
#endif
    float* zg = out + (size_t)g * MH;
    fill_zero_kernel<<<blocksMH, T, 0, stream>>>(zg, MH);
    spmm_scatter_kernel<<<blocksSp, T, 0, stream>>>(edges[g], bufA, zg, nE);
    finish_kernel<false><<<blocksMH, T, 0, stream>>>(zg, deg + M, b1, M);
  }
}